// ScaledDotProductAttention_32186484916760
// MI455X (gfx1250) — compile-verified
//
#include <hip/hip_runtime.h>
#include <hip/hip_bf16.h>

typedef __attribute__((ext_vector_type(2))) float v2f;
typedef __attribute__((ext_vector_type(8))) float v8f;

#define B_  4
#define H_  8
#define L_  1024
#define D_  64
#define SSTRIDE 1028   // 1024 + 4 pad: bank = (4*row + col) % 64 -> conflict-free column reads

// Fused attention: one block = one (b, h, 16-row q tile).
//  phase 1: S = (Q/8) K^T via v_wmma_f32_16x16x4_f32, mask + score applied, into LDS
//  phase 2: full-row softmax (wave32 shuffle reductions)
//  phase 3: write attn, banded coverage -> atomic f32 adds into new_cover
//  phase 4: O = attn * V via v_wmma_f32_16x16x4_f32
__global__ __launch_bounds__(256)
void attn_fused_kernel(const float* __restrict__ q,
                       const float* __restrict__ k,
                       const float* __restrict__ v,
                       const int*   __restrict__ mask,
                       const float* __restrict__ score,
                       float* __restrict__ outO,
                       float* __restrict__ outA,
                       float* __restrict__ outC)
{
    __shared__ float sQ[16][68];          // padded: bank = (4r + c) % 64
    __shared__ float sS[16 * SSTRIDE];    // 16 x 1024 scores/attn tile
    __shared__ float sO[2][16 * 64];      // K-split partial outputs

    const int b    = blockIdx.z;
    const int h    = blockIdx.y;
    const int qt   = blockIdx.x;
    const int tid  = threadIdx.x;
    const int wave = tid >> 5;
    const int lane = tid & 31;
    const int half = lane >> 4;           // which 16-lane half
    const int l16  = lane & 15;

    const int bh = b * H_ + h;
    const int q0 = qt * 16;

    // ---- load Q tile, pre-scaled by 1/TEMPERATURE ----
    {
        const float* qp = q + (size_t)(bh * L_ + q0) * D_;
        for (int i = tid; i < 16 * 64; i += 256) {
            int r = i >> 6, c = i & 63;
            sQ[r][c] = qp[r * 64 + c] * 0.125f;
        }
    }
    __syncthreads();

    // ---- phase 1: S = (Q/8) K^T, masked, * score_matrix ----
    {
        const float* kmat = k + (size_t)bh * L_ * D_;
        const int*   mrow = mask  + (size_t)b  * L_ * L_;
        const float* srow = score + (size_t)bh * L_ * L_;
        for (int t = 0; t < 8; ++t) {
            const int kt = wave * 8 + t;          // contiguous k-range per wave
            const int kb = kt * 16;
            v8f acc = {};
            // A frag: row = l16, K pair (4ks + 2*half, +1) ; B frag: col = l16, same K pair
            const float* arow = &sQ[l16][half * 2];
            const float* krow = kmat + (size_t)(kb + l16) * D_ + half * 2;
            #pragma unroll
            for (int ks = 0; ks < 16; ++ks) {
                v2f a, bb;
                a.x  = arow[ks * 4 + 0];
                a.y  = arow[ks * 4 + 1];
                bb.x = krow[ks * 4 + 0];
                bb.y = krow[ks * 4 + 1];
                acc = __builtin_amdgcn_wmma_f32_16x16x4_f32(
                        false, a, false, bb, (short)0, acc, false, false);
            }
            const int kcol = kb + l16;
            #pragma unroll
            for (int r = 0; r < 8; ++r) {
                const int m    = r + half * 8;    // C layout: M = r + 8*half
                const int qrow = q0 + m;
                const int   mv = mrow[(size_t)qrow * L_ + kcol];
                const float sc = srow[(size_t)qrow * L_ + kcol];
                float s = acc[r];
                s = (mv == 0) ? -1.0e9f : s;
                sS[m * SSTRIDE + kcol] = s * sc;
            }
        }
    }
    __syncthreads();

    // ---- phase 2: row softmax; wave w owns rows 2w, 2w+1 ----
    for (int rr = 0; rr < 2; ++rr) {
        const int m = wave * 2 + rr;
        float* row = &sS[m * SSTRIDE];
        float mx = -3.402823466e38f;
        for (int c = lane; c < L_; c += 32) mx = fmaxf(mx, row[c]);
        #pragma unroll
        for (int off = 16; off >= 1; off >>= 1) mx = fmaxf(mx, __shfl_xor(mx, off));
        float sum = 0.0f;
        for (int c = lane; c < L_; c += 32) {
            float e = __expf(row[c] - mx);
            row[c] = e;
            sum += e;
        }
        #pragma unroll
        for (int off = 16; off >= 1; off >>= 1) sum += __shfl_xor(sum, off);
        const float inv = 1.0f / sum;
        for (int c = lane; c < L_; c += 32) row[c] *= inv;
    }
    __syncthreads();

    // ---- phase 3a: write attn tile (coalesced) ----
    {
        float* ap = outA + (size_t)(bh * L_ + q0) * L_;
        for (int i = tid; i < 16 * L_; i += 256) {
            int r = i >> 10, c = i & (L_ - 1);
            ap[(size_t)r * L_ + c] = sS[r * SSTRIDE + c];
        }
    }
    // ---- phase 3b: coverage: sum_r min(attn[r][c-1]+attn[r][c]+attn[r][c+1], attn[r][c]) ----
    for (int c = tid; c < L_; c += 256) {
        float part = 0.0f;
        #pragma unroll
        for (int r = 0; r < 16; ++r) {
            const float* row = &sS[r * SSTRIDE];
            float a0 = (c > 0)       ? row[c - 1] : 0.0f;
            float a1 = row[c];
            float a2 = (c < L_ - 1)  ? row[c + 1] : 0.0f;
            part += fminf(a0 + a1 + a2, a1);
        }
        atomicAdd(&outC[b * L_ + c], part);   // reduces over q-tiles and heads
    }

    // ---- phase 4: O = attn * V ; wave -> (ntile = w&3, K-half = w>>2) ----
    {
        const int ntile = wave & 3;
        const int khalf = wave >> 2;
        const float* vb   = v + (size_t)bh * L_ * D_ + ntile * 16 + l16;
        const float* arow = &sS[l16 * SSTRIDE + half * 2];
        v8f oacc = {};
        for (int ks = 0; ks < 128; ++ks) {
            const int kq = khalf * 128 + ks;      // K-step of 4
            const int k4 = kq * 4 + half * 2;
            v2f a, bb;
            a.x  = arow[kq * 4 + 0];
            a.y  = arow[kq * 4 + 1];
            bb.x = vb[(size_t)(k4 + 0) * D_];
            bb.y = vb[(size_t)(k4 + 1) * D_];
            oacc = __builtin_amdgcn_wmma_f32_16x16x4_f32(
                     false, a, false, bb, (short)0, oacc, false, false);
        }
        #pragma unroll
        for (int r = 0; r < 8; ++r)
            sO[khalf][(r + half * 8) * 64 + ntile * 16 + l16] = oacc[r];
    }
    __syncthreads();
    {
        float* op = outO + (size_t)(bh * L_ + q0) * D_;
        for (int i = tid; i < 16 * 64; i += 256)
            op[i] = sO[0][i] + sO[1][i];
    }
}

__global__ void cover_init_kernel(const float* __restrict__ cover,
                                  float* __restrict__ outC, int n)
{
    int i = blockIdx.x * blockDim.x + threadIdx.x;
    if (i < n) outC[i] = cover[i];
}

extern "C" void kernel_launch(void* const* d_in, const int* in_sizes, int n_in,
                              void* d_out, int out_size, void* d_ws, size_t ws_size,
                              hipStream_t stream) {
    const float* q     = (const float*)d_in[0];
    const float* k     = (const float*)d_in[1];
    const float* v     = (const float*)d_in[2];
    const int*   mask  = (const int*)  d_in[3];
    const float* cover = (const float*)d_in[4];
    const float* score = (const float*)d_in[5];

    float* outO = (float*)d_out;                       // [B,H,L,D]
    float* outA = outO + (size_t)B_ * H_ * L_ * D_;    // [B,H,L,L]
    float* outC = outA + (size_t)B_ * H_ * L_ * L_;    // [B,L]

    // seed new_cover = cover, then blocks atomically accumulate into it
    cover_init_kernel<<<(B_ * L_ + 255) / 256, 256, 0, stream>>>(cover, outC, B_ * L_);

    dim3 grid(L_ / 16, H_, B_);
    attn_fused_kernel<<<grid, 256, 0, stream>>>(q, k, v, mask, score, outO, outA, outC);
}